// GNNDecoderV1_86260123173220
// MI455X (gfx1250) — compile-verified
//
#include <hip/hip_runtime.h>
#include <hip/hip_bf16.h>

typedef __attribute__((ext_vector_type(2))) float v2f;
typedef __attribute__((ext_vector_type(8))) float v8f;

// Problem constants (fixed by the reference).
constexpr int Bb = 8;
constexpr int Nn = 20000;
constexpr int Ee = 320000;
constexpr int Rr = Bb * Nn;          // 160000 node-rows
constexpr int HD = Rr * 64;          // floats in one (B,N,D) tensor
constexpr int NTILE = Rr / 16;       // 10000 16-row WMMA tiles
constexpr int TPW = 5;               // tiles per wave (10000 = 250 blk * 8 waves * 5)
constexpr int CHUNK = 200;           // pool stage-1 chunk (N = 100 * 200)
constexpr int NCHUNK = Nn / CHUNK;   // 100
constexpr float EPSc = 1e-5f;

// ---------------------------------------------------------------- embed
__global__ __launch_bounds__(256) void k_embed(const float* __restrict__ x,
                                               const float* __restrict__ Win,
                                               const float* __restrict__ bin,
                                               float* __restrict__ H) {
  int idx = blockIdx.x * 256 + threadIdx.x;
  if (idx >= HD) return;
  int row = idx >> 6, d = idx & 63;
  H[idx] = fmaxf(x[row] * Win[d] + bin[d], 0.0f);
}

// ---------------------------------------------------------------- zero fill
__global__ __launch_bounds__(256) void k_zero(float* __restrict__ p, int n) {
  int idx = blockIdx.x * 256 + threadIdx.x;
  if (idx < n) p[idx] = 0.0f;
}

// ---------------------------------------------------------------- degree
__global__ __launch_bounds__(256) void k_degree(const int* __restrict__ dst,
                                                float* __restrict__ deg) {
  int e = blockIdx.x * 256 + threadIdx.x;
  if (e < Ee) atomicAdd(&deg[dst[e]], 1.0f);
}

// ---------------------------------------------------------------- dual GEMM
// T = H @ Ws, G = H @ Wm   (H: Rr x 64, W: 64 x 64, row-major (in,out))
// Weights staged once per block in LDS as k-pair float2 so every B-fragment
// is one ds_load_b64; H tiles staged per wave; fp32 WMMA 16x16x4 inner loop
// is pure DS + WMMA. Next H tile is prefetched (global_prefetch_b8).
__global__ __launch_bounds__(256) void k_gemm_dual(const float* __restrict__ H,
                                                   const float* __restrict__ Ws,
                                                   const float* __restrict__ Wm,
                                                   float* __restrict__ T,
                                                   float* __restrict__ G) {
  __shared__ float ldsH[8][16 * 64];  // 32 KB: one 16x64 fp32 tile per wave
  __shared__ float ldsW[2 * 4096];    // 32 KB: WP[w][k/2][n] = {W[k][n],W[k+1][n]}
  const int wave = threadIdx.x >> 5;
  const int lane = threadIdx.x & 31;

  // Stage both weight matrices, swizzled to k-pair float2 layout.
  const float* Wmat[2] = {Ws, Wm};
  for (int w = 0; w < 2; ++w) {
    const float* W = Wmat[w];
    float* dw = ldsW + w * 4096;
#pragma unroll
    for (int i = 0; i < 16; ++i) {
      int idx = i * 256 + threadIdx.x;  // 0..4095, coalesced global read
      int k = idx >> 6, n = idx & 63;
      dw[((k >> 1) * 64 + n) * 2 + (k & 1)] = W[idx];
    }
  }

  // Fragment geometry (ISA 7.12.2):
  // A 16x4 f32: lanes 0-15 rows M, VGPR0/1 = K,K+1; lanes 16-31 = K+2,K+3.
  // B 4x16 f32: mirrored (lane = N column).
  // C/D 16x16 f32: VGPR j -> row j (lanes 0-15) / j+8 (lanes 16-31).
  const int arow = lane & 15;
  const int kofs = (lane >> 4) * 2;   // 0 or 2
  const int ncol = lane & 15;
  const int rbase = (lane >> 4) * 8;  // 0 or 8

  float* lds = ldsH[wave];
  float* Outp[2] = {T, G};
  const int group = blockIdx.x * 8 + wave;

  for (int it = 0; it < TPW; ++it) {
    const int tile = group * TPW + it;
    const int m0 = tile * 16;
    const float* srcp = H + m0 * 64;
    // Coalesced cooperative load of the 16x64 tile (1024 floats, 32/lane).
#pragma unroll
    for (int i = 0; i < 32; ++i) lds[i * 32 + lane] = srcp[i * 32 + lane];
    // Warm the next tile while this one computes (global_prefetch_b8).
    if (tile + 1 < NTILE) __builtin_prefetch(srcp + 1024 + lane * 32, 0, 1);
    __syncthreads();  // uniform: covers weight staging (it==0) + H-tile visibility

    for (int w = 0; w < 2; ++w) {
      const float* wp = ldsW + w * 4096;
      v8f a0 = {}, a1 = {}, a2 = {}, a3 = {};
      for (int k = 0; k < 64; k += 4) {
        const int kk = k + kofs;
        v2f a = *(const v2f*)(lds + arow * 64 + kk);  // ds_load_b64
        const float* wb = wp + ((kk >> 1) * 64) * 2;  // pair row in LDS
        v2f b0 = *(const v2f*)(wb + (ncol + 0) * 2);
        v2f b1 = *(const v2f*)(wb + (ncol + 16) * 2);
        v2f b2 = *(const v2f*)(wb + (ncol + 32) * 2);
        v2f b3 = *(const v2f*)(wb + (ncol + 48) * 2);
        a0 = __builtin_amdgcn_wmma_f32_16x16x4_f32(false, a, false, b0, (short)0, a0, false, false);
        a1 = __builtin_amdgcn_wmma_f32_16x16x4_f32(false, a, false, b1, (short)0, a1, false, false);
        a2 = __builtin_amdgcn_wmma_f32_16x16x4_f32(false, a, false, b2, (short)0, a2, false, false);
        a3 = __builtin_amdgcn_wmma_f32_16x16x4_f32(false, a, false, b3, (short)0, a3, false, false);
      }
      float* out = Outp[w] + m0 * 64;
#pragma unroll
      for (int j = 0; j < 8; ++j) {
        const int r = (rbase + j) * 64;
        out[r + 0 + ncol] = a0[j];
        out[r + 16 + ncol] = a1[j];
        out[r + 32 + ncol] = a2[j];
        out[r + 48 + ncol] = a3[j];
      }
    }
  }
}

// ---------------------------------------------------------------- edge scatter
// AGG[b, dst, :] += G[b, src, :]    (G = h @ Wm, computed once per node)
__global__ __launch_bounds__(256) void k_edge_scatter(const float* __restrict__ G,
                                                      const int* __restrict__ src,
                                                      const int* __restrict__ dst,
                                                      float* __restrict__ AGG) {
  int eid = blockIdx.x * 4 + (threadIdx.x >> 6);
  int d = threadIdx.x & 63;
  if (eid >= Ee) return;
  int s = src[eid], t = dst[eid];
#pragma unroll
  for (int b = 0; b < Bb; ++b) {
    float v = G[(b * Nn + s) * 64 + d];              // coalesced 256B row
    atomicAdd(&AGG[(b * Nn + t) * 64 + d], v);       // global_atomic_add_f32
  }
}

// ---------------------------------------------------------------- finalize
// H = LayerNorm(H + relu(T + AGG/deg + bias)) * gamma + beta, one wave/node.
__global__ __launch_bounds__(256) void k_node_finalize(float* __restrict__ H,
                                                       const float* __restrict__ T,
                                                       const float* __restrict__ AGG,
                                                       const float* __restrict__ deg,
                                                       const float* __restrict__ bias,
                                                       const float* __restrict__ gamma,
                                                       const float* __restrict__ beta) {
  int row = blockIdx.x * 8 + (threadIdx.x >> 5);
  int lane = threadIdx.x & 31;
  if (row >= Rr) return;
  int n = row % Nn;
  float invdeg = 1.0f / fmaxf(deg[n], 1.0f);
  int base = row * 64;
  int d0 = lane, d1 = lane + 32;
  float h0 = H[base + d0], h1 = H[base + d1];
  float z0 = h0 + fmaxf(T[base + d0] + AGG[base + d0] * invdeg + bias[d0], 0.0f);
  float z1 = h1 + fmaxf(T[base + d1] + AGG[base + d1] * invdeg + bias[d1], 0.0f);
  float s = z0 + z1;
#pragma unroll
  for (int m = 16; m >= 1; m >>= 1) s += __shfl_xor(s, m, 32);
  float mu = s * (1.0f / 64.0f);
  float e0 = z0 - mu, e1 = z1 - mu;
  float v = e0 * e0 + e1 * e1;
#pragma unroll
  for (int m = 16; m >= 1; m >>= 1) v += __shfl_xor(v, m, 32);
  float rstd = rsqrtf(v * (1.0f / 64.0f) + EPSc);
  H[base + d0] = gamma[d0] * e0 * rstd + beta[d0];
  H[base + d1] = gamma[d1] * e1 * rstd + beta[d1];
}

// ---------------------------------------------------------------- pooling
__global__ __launch_bounds__(256) void k_pool1(const float* __restrict__ H,
                                               float* __restrict__ psum,
                                               float* __restrict__ pmax) {
  __shared__ float ss[4][64], sm[4][64];
  int c = blockIdx.x, b = blockIdx.y;
  int d = threadIdx.x & 63, r0 = threadIdx.x >> 6;
  float s = 0.0f, m = -__builtin_inff();
  int base = (b * Nn + c * CHUNK) * 64;
  for (int n = r0; n < CHUNK; n += 4) {
    float v = H[base + n * 64 + d];
    s += v;
    m = fmaxf(m, v);
  }
  ss[r0][d] = s;
  sm[r0][d] = m;
  __syncthreads();
  if (r0 == 0) {
    s = ss[0][d] + ss[1][d] + ss[2][d] + ss[3][d];
    m = fmaxf(fmaxf(sm[0][d], sm[1][d]), fmaxf(sm[2][d], sm[3][d]));
    psum[(b * NCHUNK + c) * 64 + d] = s;
    pmax[(b * NCHUNK + c) * 64 + d] = m;
  }
}

__global__ __launch_bounds__(64) void k_pool2(const float* __restrict__ psum,
                                              const float* __restrict__ pmax,
                                              float* __restrict__ ge) {
  int b = blockIdx.x, d = threadIdx.x;
  float s = 0.0f, m = -__builtin_inff();
  for (int c = 0; c < NCHUNK; ++c) {
    s += psum[(b * NCHUNK + c) * 64 + d];
    m = fmaxf(m, pmax[(b * NCHUNK + c) * 64 + d]);
  }
  ge[b * 128 + d] = s / (float)Nn;
  ge[b * 128 + 64 + d] = m;
}

// ---------------------------------------------------------------- MLP head
__global__ __launch_bounds__(64) void k_head(const float* __restrict__ ge,
                                             const float* __restrict__ Wh1,
                                             const float* __restrict__ bh1,
                                             const float* __restrict__ Wh2,
                                             const float* __restrict__ bh2,
                                             float* __restrict__ out) {
  __shared__ float sm[64];
  int b = blockIdx.x, j = threadIdx.x;
  float acc = bh1[j];
  for (int i = 0; i < 128; ++i) acc += ge[b * 128 + i] * Wh1[i * 64 + j];
  sm[j] = fmaxf(acc, 0.0f) * Wh2[j];
  __syncthreads();
  if (j == 0) {
    float s = bh2[0];
    for (int i = 0; i < 64; ++i) s += sm[i];
    out[b] = s;
  }
}

// ---------------------------------------------------------------- launcher
extern "C" void kernel_launch(void* const* d_in, const int* in_sizes, int n_in,
                              void* d_out, int out_size, void* d_ws, size_t ws_size,
                              hipStream_t stream) {
  const float* x     = (const float*)d_in[0];
  const int*   ei    = (const int*)d_in[1];
  const float* Win   = (const float*)d_in[2];
  const float* bin   = (const float*)d_in[3];
  const float* Wsl[2]  = {(const float*)d_in[4], (const float*)d_in[9]};
  const float* Wml[2]  = {(const float*)d_in[5], (const float*)d_in[10]};
  const float* biasl[2]= {(const float*)d_in[6], (const float*)d_in[11]};
  const float* gl[2]   = {(const float*)d_in[7], (const float*)d_in[12]};
  const float* bel[2]  = {(const float*)d_in[8], (const float*)d_in[13]};
  const float* Wh1   = (const float*)d_in[14];
  const float* bh1   = (const float*)d_in[15];
  const float* Wh2   = (const float*)d_in[16];
  const float* bh2   = (const float*)d_in[17];
  const int* srcI = ei;
  const int* dstI = ei + Ee;

  char* ws = (char*)d_ws;
  const size_t SZ = (size_t)HD * sizeof(float);  // 40.96 MB per tensor
  float* H    = (float*)(ws);
  float* T    = (float*)(ws + SZ);
  float* G    = (float*)(ws + 2 * SZ);
  float* AGG  = (float*)(ws + 3 * SZ);
  float* deg  = (float*)(ws + 4 * SZ);
  float* psum = (float*)(ws + 4 * SZ + 81920);
  float* pmax = psum + Bb * NCHUNK * 64;
  float* ge   = pmax + Bb * NCHUNK * 64;

  k_embed<<<(HD + 255) / 256, 256, 0, stream>>>(x, Win, bin, H);
  k_zero<<<(Nn + 255) / 256, 256, 0, stream>>>(deg, Nn);
  k_degree<<<(Ee + 255) / 256, 256, 0, stream>>>(dstI, deg);

  for (int layer = 0; layer < 2; ++layer) {
    k_zero<<<(HD + 255) / 256, 256, 0, stream>>>(AGG, HD);
    k_gemm_dual<<<NTILE / (8 * TPW), 256, 0, stream>>>(H, Wsl[layer], Wml[layer], T, G);
    k_edge_scatter<<<Ee / 4, 256, 0, stream>>>(G, srcI, dstI, AGG);
    k_node_finalize<<<Rr / 8, 256, 0, stream>>>(H, T, AGG, deg, biasl[layer],
                                                gl[layer], bel[layer]);
  }

  k_pool1<<<dim3(NCHUNK, Bb), 256, 0, stream>>>(H, psum, pmax);
  k_pool2<<<Bb, 64, 0, stream>>>(psum, pmax, ge);
  k_head<<<Bb, 64, 0, stream>>>(ge, Wh1, bh1, Wh2, bh2, (float*)d_out);
}